// EmbDotSoftMax_1975684956456
// MI455X (gfx1250) — compile-verified
//
#include <hip/hip_runtime.h>

typedef __attribute__((ext_vector_type(2))) float v2f;
typedef __attribute__((ext_vector_type(8))) float v8f;

#define B_ROWS 4096
#define D_INN  1024
#define EC     600
#define N_CITY 50
#define VOCAB  40000

// ---------------------------------------------------------------------------
// Kernel 1: fill out[B, VOCAB] with 1e-6 (float4 stores, ~655 MB -> BW bound)
// ---------------------------------------------------------------------------
__global__ void __launch_bounds__(256) fill_kernel(float4* __restrict__ out, long n4) {
    long i      = (long)blockIdx.x * blockDim.x + threadIdx.x;
    long stride = (long)gridDim.x * blockDim.x;
    float4 v = make_float4(1e-6f, 1e-6f, 1e-6f, 1e-6f);
    for (; i < n4; i += stride) out[i] = v;
}

// ---------------------------------------------------------------------------
// Kernel 2: emb_pred = x @ W + b via V_WMMA_F32_16X16X4_F32
// One wave per 16x16 tile; M-tiles=256, N-tiles=38 (EC=600, last tile partial)
// A layout: lane<16 -> A[m=lane][k..k+1], lane>=16 -> A[m=lane-16][k+2..k+3]
// B layout (mirrored): lane<16 -> B[k..k+1][n=lane], lane>=16 -> B[k+2..k+3][n]
// C/D layout: vgpr r, lanes 0-15: (M=r, N=lane); lanes 16-31: (M=r+8, N=lane-16)
//
// Branch-free bounds handling: B-fragment column n only feeds D column n, and
// D columns >= EC are never stored, so for n >= EC we just CLAMP the load
// address (valid memory, garbage value) instead of predicating the loads.
// This keeps the inner loop free of EXEC save/restore.
// ---------------------------------------------------------------------------
#define NTILES_N ((EC + 15) / 16)          /* 38 */
#define NTILES_M (B_ROWS / 16)             /* 256 */
#define WAVES_PER_BLOCK 8

__global__ void __launch_bounds__(256) dense_wmma_kernel(
    const float* __restrict__ x,     // [B, D_IN]
    const float* __restrict__ W,     // [D_IN, EC]
    const float* __restrict__ bias,  // [EC]
    float* __restrict__ emb)         // [B, EC]
{
    const int wave = threadIdx.x >> 5;              // 0..7 (wave-uniform)
    const int lane = threadIdx.x & 31;
    const int tile = blockIdx.x * WAVES_PER_BLOCK + wave;
    if (tile >= NTILES_M * NTILES_N) return;        // wave-uniform exit

    const int mt = tile / NTILES_N;
    const int nt = tile % NTILES_N;
    const int m0 = mt * 16;
    const int n0 = nt * 16;

    const int hi = lane >> 4;       // 0 or 1: selects K pair {0,1} vs {2,3}
    const int lo = lane & 15;
    const int m  = m0 + lo;
    const int n  = n0 + lo;
    const bool nvalid = (n < EC);
    const int  nclamp = nvalid ? n : (EC - 1);   // in-bounds address, value unused

    v8f c = {0.f, 0.f, 0.f, 0.f, 0.f, 0.f, 0.f, 0.f};

    const float* xrow = x + (long)m * D_INN + 2 * hi;
    const float* wcol = W + (long)(2 * hi) * EC + nclamp;

    for (int k = 0; k < D_INN; k += 4) {
        // A fragment: contiguous float2 from this row of x
        const float2 av = *(const float2*)(xrow + k);
        v2f a; a.x = av.x; a.y = av.y;

        // B fragment: two consecutive K-rows of W at (clamped) column n
        v2f bm;
        bm.x = wcol[(long)k * EC];
        bm.y = wcol[(long)(k + 1) * EC];

        // 8 args: (neg_a, A, neg_b, B, c_mod, C, reuse_a, reuse_b)
        c = __builtin_amdgcn_wmma_f32_16x16x4_f32(
                false, a, false, bm, (short)0, c, false, false);
    }

    if (nvalid) {
        const float bv = bias[n];
#pragma unroll
        for (int r = 0; r < 8; ++r) {
            const int mm = m0 + r + 8 * hi;
            emb[(long)mm * EC + n] = c[r] + bv;
        }
    }
}

// ---------------------------------------------------------------------------
// Kernel 3: per-row scores (50 dots of length 600 from LDS), softmax,
// atomic scatter-add (duplicates add per tf.scatter_nd semantics).
// One 256-thread block (8 wave32s) per row; 4096 blocks.
// ---------------------------------------------------------------------------
__global__ void __launch_bounds__(256) score_scatter_kernel(
    const float* __restrict__ emb,   // [B, EC]
    const float* __restrict__ cand,  // [B, N_CITY, EC]
    const int*   __restrict__ ids,   // [B, N_CITY]
    float*       __restrict__ out)   // [B, VOCAB] (pre-filled with 1e-6)
{
    __shared__ float s_emb[EC];
    __shared__ float s_scores[N_CITY];
    __shared__ float s_p[N_CITY];

    const int row  = blockIdx.x;
    const int tid  = threadIdx.x;
    const int lane = tid & 31;
    const int wave = tid >> 5;

    for (int e = tid; e < EC; e += 256)
        s_emb[e] = emb[(long)row * EC + e];
    __syncthreads();

    // 50 candidate dots, distributed over the 8 waves; float4 streaming reads
    const float* crow = cand + (long)row * N_CITY * EC;
    for (int nn = wave; nn < N_CITY; nn += WAVES_PER_BLOCK) {
        const float4* c4 = (const float4*)(crow + (long)nn * EC);
        const float4* e4 = (const float4*)s_emb;
        float acc = 0.f;
        for (int i = lane; i < EC / 4; i += 32) {
            const float4 cv = c4[i];
            const float4 ev = e4[i];
            acc += cv.x * ev.x + cv.y * ev.y + cv.z * ev.z + cv.w * ev.w;
        }
#pragma unroll
        for (int off = 16; off > 0; off >>= 1)
            acc += __shfl_xor(acc, off, 32);
        if (lane == 0) s_scores[nn] = acc;
    }
    __syncthreads();

    // Softmax over 50 scores, done by wave 0 (wave32 shuffle reductions)
    if (wave == 0) {
        float mx = -INFINITY;
        for (int nn = lane; nn < N_CITY; nn += 32) mx = fmaxf(mx, s_scores[nn]);
#pragma unroll
        for (int off = 16; off > 0; off >>= 1)
            mx = fmaxf(mx, __shfl_xor(mx, off, 32));

        float sum = 0.f;
        for (int nn = lane; nn < N_CITY; nn += 32) {
            const float e_ = __expf(s_scores[nn] - mx);
            s_p[nn] = e_;
            sum += e_;
        }
#pragma unroll
        for (int off = 16; off > 0; off >>= 1)
            sum += __shfl_xor(sum, off, 32);

        const float inv = 1.f / sum;
        for (int nn = lane; nn < N_CITY; nn += 32) s_p[nn] *= inv;
    }
    __syncthreads();

    // Scatter: duplicates must accumulate -> atomicAdd into pre-filled row
    float* orow = out + (long)row * VOCAB;
    const int* irow = ids + (long)row * N_CITY;
    for (int nn = tid; nn < N_CITY; nn += 256)
        atomicAdd(&orow[irow[nn]], s_p[nn]);
}

// ---------------------------------------------------------------------------
extern "C" void kernel_launch(void* const* d_in, const int* in_sizes, int n_in,
                              void* d_out, int out_size, void* d_ws, size_t ws_size,
                              hipStream_t stream) {
    const float* x            = (const float*)d_in[0];
    const float* top_city_emb = (const float*)d_in[1];
    const int*   top_city_id  = (const int*)d_in[2];
    // d_in[3] = prob (unused; output buffer is d_out)
    const float* W            = (const float*)d_in[4];
    const float* b            = (const float*)d_in[5];

    float* out = (float*)d_out;
    float* emb = (float*)d_ws;   // needs 4096*600*4 = ~9.4 MB scratch

    // 1) fill output with 1e-6
    const long n4 = (long)B_ROWS * VOCAB / 4;
    fill_kernel<<<8192, 256, 0, stream>>>((float4*)out, n4);

    // 2) emb_pred = x @ W + b (WMMA f32 16x16x4)
    const int total_tiles = NTILES_M * NTILES_N;                    // 9728
    const int blocks = (total_tiles + WAVES_PER_BLOCK - 1) / WAVES_PER_BLOCK; // 1216
    dense_wmma_kernel<<<blocks, 256, 0, stream>>>(x, W, b, emb);

    // 3) scores + softmax + scatter-add
    score_scatter_kernel<<<B_ROWS, 256, 0, stream>>>(emb, top_city_emb, top_city_id, out);
}